// FAConvHeteroBlock_72095321030698
// MI455X (gfx1250) — compile-verified
//
#include <hip/hip_runtime.h>
#include <math.h>

// ---------------------------------------------------------------------------
// Types / helpers
// ---------------------------------------------------------------------------
typedef float v2f __attribute__((ext_vector_type(2)));
typedef float v8f __attribute__((ext_vector_type(8)));
typedef unsigned int u32x4 __attribute__((ext_vector_type(4)));
typedef int i32x4 __attribute__((ext_vector_type(4)));
typedef int i32x8 __attribute__((ext_vector_type(8)));

static __device__ __forceinline__ void atomAddF(float* p, float v) {
  __hip_atomic_fetch_add(p, v, __ATOMIC_RELAXED, __HIP_MEMORY_SCOPE_AGENT);
}

// ---------------------------------------------------------------------------
// Elementwise / init kernels
// ---------------------------------------------------------------------------
__global__ void zero_f32_kernel(float* __restrict__ p, int n) {
  int i = blockIdx.x * blockDim.x + threadIdx.x;
  if (i < n) p[i] = 0.0f;
}

__global__ void fill_u32_kernel(unsigned* __restrict__ p, unsigned v, int n) {
  int i = blockIdx.x * blockDim.x + threadIdx.x;
  if (i < n) p[i] = v;
}

// acc[i][c] = 0.3*x[i][c] + b1[c] + b2[c]   (3 FAConv eps terms + 2 GraphConv biases)
__global__ void acc_init_kernel(float* __restrict__ acc, const float* __restrict__ x,
                                const float* __restrict__ b1, const float* __restrict__ b2,
                                int n) {
  int i = blockIdx.x * blockDim.x + threadIdx.x;
  if (i >= n) return;
  int c = i & 127;
  acc[i] = 0.3f * x[i] + b1[c] + b2[c];
}

// C[i][c] = bias[c]
__global__ void fill_bias_kernel(float* __restrict__ C, const float* __restrict__ bias,
                                 int n, int ncols) {
  int i = blockIdx.x * blockDim.x + threadIdx.x;
  if (i < n) C[i] = bias[i % ncols];
}

// ---------------------------------------------------------------------------
// Degree / FAConv kernels
// ---------------------------------------------------------------------------
__global__ void deg_kernel(const int* __restrict__ ei, float* __restrict__ deg, int E) {
  int e = blockIdx.x * blockDim.x + threadIdx.x;
  if (e < E) atomAddF(&deg[ei[E + e]], 1.0f);
}

__global__ void dis_kernel(float* __restrict__ d, int n) {
  int i = blockIdx.x * blockDim.x + threadIdx.x;
  if (i < n) { float v = d[i]; d[i] = (v > 0.0f) ? rsqrtf(v) : 0.0f; }
}

// per-node dot products with 3 (l,r) attention vector pairs
// attL layout: [set][2][128]; s layout: [6][n]
__global__ void fa_dots_kernel(const float* __restrict__ x, const float* __restrict__ attL,
                               int s0, int s1, int s2, float* __restrict__ s, int n) {
  int i = blockIdx.x * blockDim.x + threadIdx.x;
  if (i >= n) return;
  const float* xr = x + (size_t)i * 128;
  const float* a0 = attL + s0 * 256;
  const float* a1 = attL + s1 * 256;
  const float* a2 = attL + s2 * 256;
  float d0l = 0, d0r = 0, d1l = 0, d1r = 0, d2l = 0, d2r = 0;
  for (int c = 0; c < 128; ++c) {
    float xv = xr[c];
    d0l += xv * a0[c]; d0r += xv * a0[128 + c];
    d1l += xv * a1[c]; d1r += xv * a1[128 + c];
    d2l += xv * a2[c]; d2r += xv * a2[128 + c];
  }
  s[0 * n + i] = d0l; s[1 * n + i] = d0r;
  s[2 * n + i] = d1l; s[3 * n + i] = d1r;
  s[4 * n + i] = d2l; s[5 * n + i] = d2r;
}

// wave-per-edge: acc[dst] += x[src] * tanh(sl[src]+sr[dst]) * dis[src]*dis[dst]
__global__ void fa_edge_kernel(const int* __restrict__ ei, const float* __restrict__ x,
                               const float* __restrict__ s, int pair,
                               const float* __restrict__ dis, float* __restrict__ acc,
                               int E, int n) {
  int wave = (blockIdx.x * blockDim.x + threadIdx.x) >> 5;
  int lane = threadIdx.x & 31;
  if (wave >= E) return;
  int src = ei[wave];
  int dst = ei[E + wave];
  float alpha = tanhf(s[(2 * pair) * n + src] + s[(2 * pair + 1) * n + dst]) *
                dis[src] * dis[dst];
  const float* xs = x + (size_t)src * 128;
  float* ad = acc + (size_t)dst * 128;
#pragma unroll
  for (int c = lane; c < 128; c += 32) atomAddF(ad + c, xs[c] * alpha);
}

// wave-per-edge: agg[dst] += xsrc[src]
__global__ void scatter_rows_kernel(const int* __restrict__ ei, const float* __restrict__ xsrc,
                                    float* __restrict__ agg, int E) {
  int wave = (blockIdx.x * blockDim.x + threadIdx.x) >> 5;
  int lane = threadIdx.x & 31;
  if (wave >= E) return;
  int src = ei[wave];
  int dst = ei[E + wave];
  const float* xs = xsrc + (size_t)src * 128;
  float* ad = agg + (size_t)dst * 128;
#pragma unroll
  for (int c = lane; c < 128; c += 32) atomAddF(ad + c, xs[c]);
}

// ---------------------------------------------------------------------------
// WMMA fp32 GEMM: C[M,N] += A[M,128] @ W[128,N]   (K fixed = 128)
// 256-thread block (8 waves). Block computes a 256-row x 32-col output panel.
// The shared W[:, n0:n0+32] panel (128x32 f32 = 16 KB) is staged into LDS once
// per block by the Tensor Data Mover (tensor_load_to_lds + s_wait_tensorcnt),
// then all 8 waves read B fragments from LDS. A fragments are per-wave
// global_load_b64. Each wave: 2x2 fragments of V_WMMA_F32_16X16X4_F32.
// ---------------------------------------------------------------------------
__global__ __launch_bounds__(256) void gemm_acc_kernel(const float* __restrict__ A,
                                                       const float* __restrict__ W,
                                                       float* __restrict__ C, int M, int N) {
  __shared__ float sW[128 * 32]; // row k -> sW[k*32 + (n-n0)]
  const int tid = threadIdx.x;
  const int lane = tid & 31;
  const int wid = tid >> 5;
  const int n0 = blockIdx.x * 32;
  const int m0 = blockIdx.y * 256 + wid * 32;

#if __has_builtin(__builtin_amdgcn_tensor_load_to_lds) && __has_builtin(__builtin_amdgcn_s_wait_tensorcnt)
  if (wid == 0) {
    // ---- Tensor DMA descriptor (D#), ISA cdna5 sec 8.3/8.4 ----
    unsigned ldsoff = (unsigned)(size_t)(&sW[0]);                 // LDS byte address
    unsigned long long ga = (unsigned long long)(const void*)(W + n0); // tile start
    unsigned td0 = (unsigned)N;   // tensor_dim0 (row length, elements)
    unsigned td1 = 128u;          // tensor_dim1 (rows)
    unsigned t0 = 32u;            // tile_dim0
    unsigned t1 = 128u;           // tile_dim1
    unsigned long long st0 = (unsigned long long)N; // tensor_dim0_stride
    u32x4 g0;
    g0[0] = 1u;                                            // count=1, user mode
    g0[1] = ldsoff;                                        // lds_addr
    g0[2] = (unsigned)(ga & 0xFFFFFFFFu);                  // global_addr[31:0]
    g0[3] = (unsigned)((ga >> 32) & 0x1FFFFFFu) | (2u << 30); // addr[56:32] | type=2
    i32x8 g1;
    g1[0] = (int)0x20000u;                                 // data_size=2 (4 bytes)
    g1[1] = (int)((td0 & 0xFFFFu) << 16);                  // tensor_dim0[15:0]
    g1[2] = (int)((td0 >> 16) | ((td1 & 0xFFFFu) << 16));  // td0[31:16] | td1[15:0]
    g1[3] = (int)((td1 >> 16) | (t0 << 16));               // td1[31:16] | tile_dim0
    g1[4] = (int)t1;                                       // tile_dim1 | tile_dim2=0
    g1[5] = (int)(st0 & 0xFFFFFFFFu);                      // stride0[31:0]
    g1[6] = (int)((st0 >> 32) & 0xFFFFu);                  // stride0[47:32] | stride1[15:0]=0
    g1[7] = 0;                                             // stride1[47:16]=0
    i32x4 gz4 = {0, 0, 0, 0};                              // groups 2/3 unused (2D tile)
    i32x8 gz8 = {0, 0, 0, 0, 0, 0, 0, 0};                  // trailing group (unused)
    __builtin_amdgcn_tensor_load_to_lds(g0, g1, gz4, gz4, gz8, 0);
    __builtin_amdgcn_s_wait_tensorcnt(0);
  }
  __syncthreads();
#else
  for (int i = tid; i < 128 * 32; i += 256) {
    int r = i >> 5, cidx = i & 31;
    sW[i] = W[(size_t)r * N + n0 + cidx];
  }
  __syncthreads();
#endif

  const int lh = lane & 15;
  const int kh = (lane >> 4) << 1;   // 0 or 2
  const int rsel = (lane >> 4) << 3; // 0 or 8

  v8f c00 = {}, c01 = {}, c10 = {}, c11 = {};
  {
    int colA = n0 + lh, colB = n0 + 16 + lh;
    int rb = m0 + rsel;
#pragma unroll
    for (int v = 0; v < 8; ++v) {
      c00[v] = C[(size_t)(rb + v) * N + colA];
      c01[v] = C[(size_t)(rb + v) * N + colB];
      c10[v] = C[(size_t)(rb + 16 + v) * N + colA];
      c11[v] = C[(size_t)(rb + 16 + v) * N + colB];
    }
  }

  const float* arow0 = A + (size_t)(m0 + lh) * 128;
  const float* arow1 = A + (size_t)(m0 + 16 + lh) * 128;
  for (int k0 = 0; k0 < 128; k0 += 4) {
    int kb = k0 + kh;
    v2f a0, a1, b0, b1;
    a0.x = arow0[kb];     a0.y = arow0[kb + 1];
    a1.x = arow1[kb];     a1.y = arow1[kb + 1];
    const float* sr0 = sW + kb * 32;
    const float* sr1 = sW + (kb + 1) * 32;
    b0.x = sr0[lh];       b0.y = sr1[lh];
    b1.x = sr0[16 + lh];  b1.y = sr1[16 + lh];
    c00 = __builtin_amdgcn_wmma_f32_16x16x4_f32(false, a0, false, b0, (short)0, c00, false, false);
    c01 = __builtin_amdgcn_wmma_f32_16x16x4_f32(false, a0, false, b1, (short)0, c01, false, false);
    c10 = __builtin_amdgcn_wmma_f32_16x16x4_f32(false, a1, false, b0, (short)0, c10, false, false);
    c11 = __builtin_amdgcn_wmma_f32_16x16x4_f32(false, a1, false, b1, (short)0, c11, false, false);
  }

  {
    int colA = n0 + lh, colB = n0 + 16 + lh;
    int rb = m0 + rsel;
#pragma unroll
    for (int v = 0; v < 8; ++v) {
      C[(size_t)(rb + v) * N + colA]      = c00[v];
      C[(size_t)(rb + v) * N + colB]      = c01[v];
      C[(size_t)(rb + 16 + v) * N + colA] = c10[v];
      C[(size_t)(rb + 16 + v) * N + colB] = c11[v];
    }
  }
}

// ---------------------------------------------------------------------------
// PairNorm / BatchNorm stats
// ---------------------------------------------------------------------------
// one block per column c (128 blocks), 256 threads stride over rows
__global__ void colstats_kernel(const float* __restrict__ a, float scale, int rows,
                                float* __restrict__ colsum, float* __restrict__ colsumsq) {
  __shared__ float ssum[256], ssq[256];
  int c = blockIdx.x;
  float s = 0, q = 0;
  for (int i = threadIdx.x; i < rows; i += blockDim.x) {
    float v = a[(size_t)i * 128 + c] * scale;
    s += v; q += v * v;
  }
  ssum[threadIdx.x] = s; ssq[threadIdx.x] = q;
  __syncthreads();
  for (int st = 128; st > 0; st >>= 1) {
    if (threadIdx.x < st) {
      ssum[threadIdx.x] += ssum[threadIdx.x + st];
      ssq[threadIdx.x]  += ssq[threadIdx.x + st];
    }
    __syncthreads();
  }
  if (threadIdx.x == 0) { colsum[c] = ssum[0]; colsumsq[c] = ssq[0]; }
}

// single block, 128 threads: mu[c], inv = rsqrt(mean_i(sum_c (y-mu)^2) + 1e-6)
__global__ void pn_finalize_kernel(const float* __restrict__ colsum,
                                   const float* __restrict__ colsumsq, int rows,
                                   float* __restrict__ mu, float* __restrict__ inv) {
  __shared__ float sh[128];
  int c = threadIdx.x;
  float m = colsum[c] / (float)rows;
  mu[c] = m;
  sh[c] = colsumsq[c] - (float)rows * m * m;
  __syncthreads();
  for (int st = 64; st > 0; st >>= 1) {
    if (c < st) sh[c] += sh[c + st];
    __syncthreads();
  }
  if (c == 0) inv[0] = rsqrtf(sh[0] / (float)rows + 1e-6f);
}

// x = relu((acc*0.2 - mu[c]) * inv)
__global__ void pn_apply_kernel(const float* __restrict__ acc, const float* __restrict__ mu,
                                const float* __restrict__ inv, float* __restrict__ x, int n) {
  int i = blockIdx.x * blockDim.x + threadIdx.x;
  if (i >= n) return;
  float y = (acc[i] * 0.2f - mu[i & 127]) * inv[0];
  x[i] = fmaxf(0.0f, y);
}

// out = leaky_relu((h - mu)*rsqrt(var+1e-5)*gamma + beta)
__global__ void bn_apply_kernel(const float* __restrict__ h, const float* __restrict__ colsum,
                                const float* __restrict__ colsumsq, const float* __restrict__ gamma,
                                const float* __restrict__ beta, float* __restrict__ out,
                                int rows, int n) {
  int i = blockIdx.x * blockDim.x + threadIdx.x;
  if (i >= n) return;
  int c = i & 127;
  float mu = colsum[c] / (float)rows;
  float var = colsumsq[c] / (float)rows - mu * mu;
  float y = (h[i] - mu) * rsqrtf(var + 1e-5f) * gamma[c] + beta[c];
  out[i] = (y > 0.0f) ? y : 0.01f * y;
}

// ---------------------------------------------------------------------------
// TransformerConv kernels
// ---------------------------------------------------------------------------
__global__ void offset_edges_kernel(const int* __restrict__ ei, int so, int dof,
                                    int* __restrict__ hsrc, int* __restrict__ hdst,
                                    int E, int base) {
  int e = blockIdx.x * blockDim.x + threadIdx.x;
  if (e >= E) return;
  hsrc[base + e] = ei[e] + so;
  hdst[base + e] = ei[E + e] + dof;
}

// wave-per-edge: per-head dot(q[dst],k[src]) * 1/sqrt(128); ordered-uint atomic max
__global__ void attn_logit_kernel(const int* __restrict__ hsrc, const int* __restrict__ hdst,
                                  const float* __restrict__ q, const float* __restrict__ k,
                                  float* __restrict__ logit, unsigned* __restrict__ mkey,
                                  int Etot) {
  int wave = (blockIdx.x * blockDim.x + threadIdx.x) >> 5;
  int lane = threadIdx.x & 31;
  if (wave >= Etot) return;
  int src = hsrc[wave], dst = hdst[wave];
  const float scale = 0.08838834764831845f; // 1/sqrt(128)
#pragma unroll
  for (int h = 0; h < 2; ++h) {
    const float* qp = q + (size_t)dst * 256 + h * 128;
    const float* kp = k + (size_t)src * 256 + h * 128;
    float p = 0.0f;
#pragma unroll
    for (int c = lane; c < 128; c += 32) p += qp[c] * kp[c];
#pragma unroll
    for (int off = 16; off > 0; off >>= 1) p += __shfl_xor(p, off, 32);
    if (lane == 0) {
      float lg = p * scale;
      logit[(size_t)wave * 2 + h] = lg;
      int ib = __float_as_int(lg);
      unsigned key = (ib < 0) ? ~(unsigned)ib : ((unsigned)ib | 0x80000000u);
      __hip_atomic_fetch_max(&mkey[dst * 2 + h], key, __ATOMIC_RELAXED,
                             __HIP_MEMORY_SCOPE_AGENT);
    }
  }
}

// thread per (edge,head): p = exp(logit - m[dst]); denom[dst] += p (overwrites logit buf)
__global__ void attn_p_kernel(const int* __restrict__ hdst, float* __restrict__ logit_p,
                              const unsigned* __restrict__ mkey, float* __restrict__ denom,
                              int Etot2) {
  int idx = blockIdx.x * blockDim.x + threadIdx.x;
  if (idx >= Etot2) return;
  int e = idx >> 1, h = idx & 1;
  int dst = hdst[e];
  unsigned key = mkey[dst * 2 + h];
  int ib = (key & 0x80000000u) ? (int)(key ^ 0x80000000u) : (int)~key;
  float m = __int_as_float(ib);
  float p = expf(logit_p[idx] - m);
  logit_p[idx] = p;
  atomAddF(&denom[dst * 2 + h], p);
}

// wave-per-edge: hb[dst] += 0.5 * alpha_h * v[src][h]   (head mean fused)
__global__ void attn_agg_kernel(const int* __restrict__ hsrc, const int* __restrict__ hdst,
                                const float* __restrict__ p, const float* __restrict__ denom,
                                const float* __restrict__ v, float* __restrict__ hb, int Etot) {
  int wave = (blockIdx.x * blockDim.x + threadIdx.x) >> 5;
  int lane = threadIdx.x & 31;
  if (wave >= Etot) return;
  int src = hsrc[wave], dst = hdst[wave];
#pragma unroll
  for (int h = 0; h < 2; ++h) {
    float alpha = 0.5f * p[(size_t)wave * 2 + h] / denom[dst * 2 + h];
    const float* vp = v + (size_t)src * 256 + h * 128;
    float* hp = hb + (size_t)dst * 128;
#pragma unroll
    for (int c = lane; c < 128; c += 32) atomAddF(hp + c, vp[c] * alpha);
  }
}

// ---------------------------------------------------------------------------
// Host orchestration
// ---------------------------------------------------------------------------
static inline int cdiv(int a, int b) { return (a + b - 1) / b; }

extern "C" void kernel_launch(void* const* d_in, const int* in_sizes, int n_in,
                              void* d_out, int out_size, void* d_ws, size_t ws_size,
                              hipStream_t stream) {
  const int N = 8192, CH = 128, L = 2;
  const int NH = 3 * N; // homo nodes

  const float* x_in[3] = { (const float*)d_in[0], (const float*)d_in[1],
                           (const float*)d_in[2] }; // audio, visual, text
  const int* ei[15];
  int Es[15];
  int Etot = 0;
  for (int t = 0; t < 15; ++t) {
    ei[t] = (const int*)d_in[3 + t];
    Es[t] = in_sizes[3 + t] / 2;
    Etot += Es[t];
  }
  const float* fa_att = (const float*)d_in[18]; // [L][9][2][128]
  const float* gcw    = (const float*)d_in[19]; // [L][6][2][128][128]
  const float* gcb    = (const float*)d_in[20]; // [L][6][128]
  const float* wq = (const float*)d_in[21]; const float* bq = (const float*)d_in[22];
  const float* wk = (const float*)d_in[23]; const float* bk = (const float*)d_in[24];
  const float* wv = (const float*)d_in[25]; const float* bv = (const float*)d_in[26];
  const float* wsk = (const float*)d_in[27]; const float* bsk = (const float*)d_in[28];
  const float* gamma = (const float*)d_in[29]; const float* beta = (const float*)d_in[30];
  float* out = (float*)d_out;

  // ---- workspace carve (bump allocator, 256B aligned) ----
  char* base = (char*)d_ws;
  size_t off = 0;
  auto alloc = [&](size_t nbytes) -> void* {
    off = (off + 255) & ~(size_t)255;
    void* p = base + off;
    off += nbytes;
    return p;
  };
  float* X[3];   for (int m = 0; m < 3; ++m) X[m]   = (float*)alloc((size_t)N * CH * 4);
  float* ACC[3]; for (int m = 0; m < 3; ++m) ACC[m] = (float*)alloc((size_t)N * CH * 4);
  float* AGG  = (float*)alloc((size_t)N * CH * 4);
  float* DIS  = (float*)alloc((size_t)9 * N * 4);
  float* S    = (float*)alloc((size_t)6 * N * 4);
  float* HOMO = (float*)alloc((size_t)NH * CH * 4);
  float* Q    = (float*)alloc((size_t)NH * 256 * 4);
  float* Kb   = (float*)alloc((size_t)NH * 256 * 4);
  float* Vb   = (float*)alloc((size_t)NH * 256 * 4);
  int*   HS   = (int*)alloc((size_t)Etot * 4);
  int*   HDs  = (int*)alloc((size_t)Etot * 4);
  float* P    = (float*)alloc((size_t)Etot * 2 * 4);
  unsigned* MK = (unsigned*)alloc((size_t)NH * 2 * 4);
  float* DEN  = (float*)alloc((size_t)NH * 2 * 4);
  float* HB   = (float*)alloc((size_t)NH * CH * 4);
  float* CS   = (float*)alloc(CH * 4);
  float* CQ   = (float*)alloc(CH * 4);
  float* MU   = (float*)alloc(CH * 4);
  float* INV  = (float*)alloc(4);
  (void)ws_size; (void)n_in; (void)out_size;

  const int TB = 256;
  const int NE = N * CH; // elements per modality matrix

  // ---- stage inputs into mutable x buffers ----
  for (int m = 0; m < 3; ++m)
    (void)hipMemcpyAsync(X[m], x_in[m], (size_t)NE * 4, hipMemcpyDeviceToDevice, stream);

  // ---- degrees -> d^-1/2 for the 9 FA edge sets (set s uses input t = s) ----
  for (int s = 0; s < 9; ++s) {
    zero_f32_kernel<<<cdiv(N, TB), TB, 0, stream>>>(DIS + s * N, N);
    deg_kernel<<<cdiv(Es[s], TB), TB, 0, stream>>>(ei[s], DIS + s * N, Es[s]);
    dis_kernel<<<cdiv(N, TB), TB, 0, stream>>>(DIS + s * N, N);
  }

  // ---- build homogeneous edge list (order must match reference `offs`) ----
  {
    const int oa = 0, ot = N, ov = 2 * N;
    const int so[15]  = { oa, ov, ot, oa, ov, ot, oa, ov, ot, oa, oa, ov, ov, ot, ot };
    const int dof[15] = { oa, ov, ot, oa, ov, ot, oa, ov, ot, ov, ot, oa, ot, oa, ov };
    int eb = 0;
    for (int t = 0; t < 15; ++t) {
      offset_edges_kernel<<<cdiv(Es[t], TB), TB, 0, stream>>>(ei[t], so[t], dof[t],
                                                              HS, HDs, Es[t], eb);
      eb += Es[t];
    }
  }

  // ---- GraphConv config per destination modality ----
  // {edge input idx, gc set, src modality}
  const int gc_cfg[3][2][3] = {
    { {11, 2, 1}, {13, 4, 2} },  // audio  <- va_cross(visual), ta_cross(text)
    { { 9, 0, 0}, {14, 5, 2} },  // visual <- av_cross(audio),  tv_cross(text)
    { {10, 1, 0}, {12, 3, 1} },  // text   <- at_cross(audio),  vt_cross(visual)
  };

  // ---- layer loop ----
  for (int l = 0; l < L; ++l) {
    const float* attL = fa_att + (size_t)l * 9 * 2 * CH;
    for (int m = 0; m < 3; ++m) {
      int setA = gc_cfg[m][0][1], setB = gc_cfg[m][1][1];
      const float* b1 = gcb + ((size_t)l * 6 + setA) * CH;
      const float* b2 = gcb + ((size_t)l * 6 + setB) * CH;
      acc_init_kernel<<<cdiv(NE, TB), TB, 0, stream>>>(ACC[m], X[m], b1, b2, NE);

      // FAConv: sets m, m+3, m+6
      fa_dots_kernel<<<cdiv(N, TB), TB, 0, stream>>>(X[m], attL, m, m + 3, m + 6, S, N);
      for (int j = 0; j < 3; ++j) {
        int s = m + 3 * j;
        fa_edge_kernel<<<cdiv(Es[s] * 32, TB), TB, 0, stream>>>(
            ei[s], X[m], S, j, DIS + s * N, ACC[m], Es[s], N);
      }

      // GraphConv x2: acc += agg@W_rel + x@W_root (WMMA fp32 GEMMs, TDM-staged W)
      for (int g = 0; g < 2; ++g) {
        int t = gc_cfg[m][g][0], set = gc_cfg[m][g][1], sm = gc_cfg[m][g][2];
        zero_f32_kernel<<<cdiv(NE, TB), TB, 0, stream>>>(AGG, NE);
        scatter_rows_kernel<<<cdiv(Es[t] * 32, TB), TB, 0, stream>>>(ei[t], X[sm], AGG, Es[t]);
        const float* w_rel  = gcw + (((size_t)l * 6 + set) * 2 + 0) * CH * CH;
        const float* w_root = gcw + (((size_t)l * 6 + set) * 2 + 1) * CH * CH;
        gemm_acc_kernel<<<dim3(CH / 32, N / 256), 256, 0, stream>>>(AGG,  w_rel,  ACC[m], N, CH);
        gemm_acc_kernel<<<dim3(CH / 32, N / 256), 256, 0, stream>>>(X[m], w_root, ACC[m], N, CH);
      }
    }
    // PairNorm + ReLU (after all ACCs so X reads are complete)
    for (int m = 0; m < 3; ++m) {
      colstats_kernel<<<CH, 256, 0, stream>>>(ACC[m], 0.2f, N, CS, CQ);
      pn_finalize_kernel<<<1, CH, 0, stream>>>(CS, CQ, N, MU, INV);
      pn_apply_kernel<<<cdiv(NE, TB), TB, 0, stream>>>(ACC[m], MU, INV, X[m], NE);
    }
  }

  // ---- homo concat: audio, text, visual ----
  (void)hipMemcpyAsync(HOMO,                X[0], (size_t)NE * 4, hipMemcpyDeviceToDevice, stream);
  (void)hipMemcpyAsync(HOMO + (size_t)NE,   X[2], (size_t)NE * 4, hipMemcpyDeviceToDevice, stream);
  (void)hipMemcpyAsync(HOMO + (size_t)2*NE, X[1], (size_t)NE * 4, hipMemcpyDeviceToDevice, stream);

  // ---- Q/K/V/skip projections (bias init + WMMA GEMM) ----
  fill_bias_kernel<<<cdiv(NH * 256, TB), TB, 0, stream>>>(Q,  bq,  NH * 256, 256);
  fill_bias_kernel<<<cdiv(NH * 256, TB), TB, 0, stream>>>(Kb, bk,  NH * 256, 256);
  fill_bias_kernel<<<cdiv(NH * 256, TB), TB, 0, stream>>>(Vb, bv,  NH * 256, 256);
  fill_bias_kernel<<<cdiv(NH * CH,  TB), TB, 0, stream>>>(HB, bsk, NH * CH,  CH);
  gemm_acc_kernel<<<dim3(256 / 32, NH / 256), 256, 0, stream>>>(HOMO, wq,  Q,  NH, 256);
  gemm_acc_kernel<<<dim3(256 / 32, NH / 256), 256, 0, stream>>>(HOMO, wk,  Kb, NH, 256);
  gemm_acc_kernel<<<dim3(256 / 32, NH / 256), 256, 0, stream>>>(HOMO, wv,  Vb, NH, 256);
  gemm_acc_kernel<<<dim3(CH / 32,  NH / 256), 256, 0, stream>>>(HOMO, wsk, HB, NH, CH);

  // ---- edge softmax + aggregation ----
  fill_u32_kernel<<<cdiv(NH * 2, TB), TB, 0, stream>>>(MK, 0x007FFFFFu, NH * 2); // enc(-inf)
  zero_f32_kernel<<<cdiv(NH * 2, TB), TB, 0, stream>>>(DEN, NH * 2);
  attn_logit_kernel<<<cdiv(Etot * 32, TB), TB, 0, stream>>>(HS, HDs, Q, Kb, P, MK, Etot);
  attn_p_kernel<<<cdiv(Etot * 2, TB), TB, 0, stream>>>(HDs, P, MK, DEN, Etot * 2);
  attn_agg_kernel<<<cdiv(Etot * 32, TB), TB, 0, stream>>>(HS, HDs, P, DEN, Vb, HB, Etot);

  // ---- BatchNorm (train stats) + leaky ReLU -> d_out ----
  colstats_kernel<<<CH, 256, 0, stream>>>(HB, 1.0f, NH, CS, CQ);
  bn_apply_kernel<<<cdiv(NH * CH, TB), TB, 0, stream>>>(HB, CS, CQ, gamma, beta, out,
                                                        NH, NH * CH);
}